// VectorQuantizerEMA_30176440222158
// MI455X (gfx1250) — compile-verified
//
#include <hip/hip_runtime.h>
#include <hip/hip_bf16.h>

typedef __attribute__((ext_vector_type(2))) float v2f;
typedef __attribute__((ext_vector_type(8))) float v8f;

// Problem constants
#define KC   1024          // num_embeddings
#define DD   256           // embedding_dim
#define NR   32768         // B*H*W
#define HW   1024          // H*W
#define DHW  (DD*HW)       // D*H*W
#define COMMIT   0.25f
#define DECAY    0.99f
#define LAMBDA   1e-5f
#define TAU      0.5f

// Output offsets (float elements, concatenated tuple in return order)
#define OUT_LOSS      ((size_t)0)
#define OUT_QUANT     ((size_t)1)
#define OUT_PERP      ((size_t)8388609)
#define OUT_ENC       ((size_t)8388610)
#define OUT_NEWCOUNT  ((size_t)41943042)
#define OUT_NEWEMA    ((size_t)41944066)
#define OUT_NEWEMB    ((size_t)42206210)

// Workspace layout (4-byte units)
#define WS_XNORM     0        // NR floats
#define WS_ENORM     32768    // KC floats
#define WS_COUNTK    33792    // KC floats (zeroed each call)
#define WS_NEWCOUNT  34816    // KC floats
#define WS_LOSS      35840    // 1 float   (zeroed each call)
#define WS_HIST      36864    // KC ints   (zeroed each call)
#define WS_IDX       37888    // NR ints
// total ~276 KB

// x[n,d] lives at inputs[b, d, h, w]; n = ((b*H+h)*W+w)
__device__ __forceinline__ size_t xoff(int n, int d) {
    return (size_t)(n >> 10) * (size_t)DHW + (size_t)d * (size_t)HW + (size_t)(n & 1023);
}

// issue one per-lane 16B async DMA: global -> LDS (ASYNCcnt tracked)
__device__ __forceinline__ void async_g2l_b128(unsigned lds_byte_addr, const float* gaddr) {
    asm volatile("global_load_async_to_lds_b128 %0, %1, off"
                 :: "v"(lds_byte_addr), "v"(gaddr) : "memory");
}
__device__ __forceinline__ void wait_async0() {
    asm volatile("s_wait_asynccnt 0x0" ::: "memory");
}

// ---------------------------------------------------------------- zero ws
__global__ void zero_kernel(float* wsf, int* wsi) {
    int t = blockIdx.x * blockDim.x + threadIdx.x;
    if (t < KC) { wsf[WS_COUNTK + t] = 0.0f; wsi[WS_HIST + t] = 0; }
    if (t == 0) wsf[WS_LOSS] = 0.0f;
}

// ---------------------------------------------------------------- row norms
__global__ void norms_kernel(const float* __restrict__ inp,
                             const float* __restrict__ weight,
                             float* wsf) {
    int t = blockIdx.x * blockDim.x + threadIdx.x;
    if (t < NR) {
        float s = 0.0f;
        size_t base = (size_t)(t >> 10) * (size_t)DHW + (size_t)(t & 1023);
        for (int d = 0; d < DD; ++d) { float v = inp[base + (size_t)d * HW]; s += v * v; }
        wsf[WS_XNORM + t] = s;
    }
    if (t < KC) {
        float s = 0.0f;
        const float* w = weight + (size_t)t * DD;
        for (int d = 0; d < DD; ++d) { float v = w[d]; s += v * v; }
        wsf[WS_ENORM + t] = s;
    }
}

// ---------------------------------------------------------------- distances GEMM (WMMA f32)
// block = 8 waves sharing one 64-wide k tile; weight tile staged in LDS via async DMA.
// each wave: 16 rows (n) x 64 cols (k); D-loop in steps of 4.
#define WPAD 260   // LDS row stride (dwords): banks = dA + 4*lo (+2 hi) mod 64, conflict-free
__global__ void __launch_bounds__(256) dist_kernel(const float* __restrict__ inp,
                                                   const float* __restrict__ weight,
                                                   const float* __restrict__ wsf,
                                                   float* __restrict__ distm) {
    __shared__ float s_w[64 * WPAD];
    int t  = threadIdx.x;
    int kb = blockIdx.x & 15;       // 16 k-blocks of 64
    int nb = blockIdx.x >> 4;       // 256 n-blocks of 128
    int k0 = kb * 64;

    // ---- stage weight[k0..k0+63][0..255] into LDS (async DMA, 16B per lane per op)
    {
        int r  = t >> 2;                         // k row 0..63
        int cq = t & 3;                          // quarter of the 256-wide row
        const float* gsrc = weight + (size_t)(k0 + r) * DD + cq * 64;
        unsigned lbase = (unsigned)(unsigned long long)&s_w[r * WPAD + cq * 64];
#pragma unroll
        for (int j = 0; j < 16; ++j)
            async_g2l_b128(lbase + j * 16, gsrc + j * 4);
        wait_async0();
    }
    __syncthreads();

    int lane = t & 31;
    int lo = lane & 15, hi = lane >> 4;
    int n0 = nb * 128 + (t >> 5) * 16;
    int n_a = n0 + lo;

    v8f c0 = {}, c1 = {}, c2 = {}, c3 = {};
    for (int d0 = 0; d0 < DD; d0 += 4) {
        int dA = d0 + 2 * hi;                    // K-dim index = v + 2*hi
        v2f a;
        a.x = inp[xoff(n_a, dA)];
        a.y = inp[xoff(n_a, dA + 1)];
        v2f b0, b1, b2, b3;
        int bi = lo * WPAD + dA;
        b0.x = s_w[bi];             b0.y = s_w[bi + 1];
        b1.x = s_w[bi + 16*WPAD];   b1.y = s_w[bi + 16*WPAD + 1];
        b2.x = s_w[bi + 32*WPAD];   b2.y = s_w[bi + 32*WPAD + 1];
        b3.x = s_w[bi + 48*WPAD];   b3.y = s_w[bi + 48*WPAD + 1];
        c0 = __builtin_amdgcn_wmma_f32_16x16x4_f32(false, a, false, b0, (short)0, c0, false, false);
        c1 = __builtin_amdgcn_wmma_f32_16x16x4_f32(false, a, false, b1, (short)0, c1, false, false);
        c2 = __builtin_amdgcn_wmma_f32_16x16x4_f32(false, a, false, b2, (short)0, c2, false, false);
        c3 = __builtin_amdgcn_wmma_f32_16x16x4_f32(false, a, false, b3, (short)0, c3, false, false);
    }
#pragma unroll
    for (int r = 0; r < 8; ++r) {
        int n = n0 + r + 8 * hi;
        float xn = wsf[WS_XNORM + n];
        int kb2 = k0 + lo;
        size_t rowo = (size_t)n * KC;
        distm[rowo + kb2]      = xn + wsf[WS_ENORM + kb2]      - 2.0f * c0[r];
        distm[rowo + kb2 + 16] = xn + wsf[WS_ENORM + kb2 + 16] - 2.0f * c1[r];
        distm[rowo + kb2 + 32] = xn + wsf[WS_ENORM + kb2 + 32] - 2.0f * c2[r];
        distm[rowo + kb2 + 48] = xn + wsf[WS_ENORM + kb2 + 48] - 2.0f * c3[r];
    }
}

// ---------------------------------------------------------------- per-row softmax / argmax / loss
// one wave per row; 8 waves per block; LDS accumulation of per-k soft counts
__global__ void __launch_bounds__(256) row_kernel(float* __restrict__ distm,       // in: dist, out: soft
                                                  const float* __restrict__ gumbel,
                                                  float* wsf, int* wsi) {
    __shared__ float s_cnt[KC];
    __shared__ float s_loss;
    for (int i = threadIdx.x; i < KC; i += blockDim.x) s_cnt[i] = 0.0f;
    if (threadIdx.x == 0) s_loss = 0.0f;
    __syncthreads();

    int lane = threadIdx.x & 31;
    int n = blockIdx.x * 8 + (threadIdx.x >> 5);
    float* drow = distm + (size_t)n * KC;
    const float* grow = gumbel + (size_t)n * KC;

    float z[32];
    float m = -3.0e38f;
    float best = -3.0e38f; int bestk = 0;
    for (int j = 0; j < 32; ++j) {
        int k = lane + 32 * j;
        float d = drow[k];
        float zz = -d * (1.0f / TAU);
        z[j] = zz;
        m = fmaxf(m, zz);
        float sc = grow[k] - d;
        if (sc > best) { best = sc; bestk = k; }
    }
    for (int off = 16; off; off >>= 1) m = fmaxf(m, __shfl_xor(m, off, 32));
    for (int off = 16; off; off >>= 1) {
        float ob = __shfl_xor(best, off, 32);
        int   ok = __shfl_xor(bestk, off, 32);
        if (ob > best || (ob == best && ok < bestk)) { best = ob; bestk = ok; }
    }
    float S = 0.0f, T = 0.0f;
    for (int j = 0; j < 32; ++j) {
        float e = expf(z[j] - m);
        T += z[j] * e;
        S += e;
        z[j] = e;
    }
    for (int off = 16; off; off >>= 1) { S += __shfl_xor(S, off, 32); T += __shfl_xor(T, off, 32); }
    float invS = 1.0f / S;
    for (int j = 0; j < 32; ++j) {
        int k = lane + 32 * j;
        float p = z[j] * invS;
        drow[k] = p;                                // soft encodings, in place (L2 resident)
        atomicAdd(&s_cnt[k], p);
    }
    if (lane == 0) {
        atomicAdd(&s_loss, T * invS - m - logf(S)); // sum_k p log p for this row
        wsi[WS_IDX + n] = bestk;
        atomicAdd(&wsi[WS_HIST + bestk], 1);
    }
    __syncthreads();
    for (int i = threadIdx.x; i < KC; i += blockDim.x)
        atomicAdd(&wsf[WS_COUNTK + i], s_cnt[i]);
    if (threadIdx.x == 0) atomicAdd(&wsf[WS_LOSS], s_loss);
}

// ---------------------------------------------------------------- small finalize (1 block)
__global__ void __launch_bounds__(1024) finalize_kernel(const float* __restrict__ ema_count,
                                                        float* wsf, const int* wsi,
                                                        float* __restrict__ out) {
    __shared__ float red[32];
    int t = threadIdx.x;
    float cnt = ema_count[t] * DECAY + wsf[WS_COUNTK + t] * (1.0f - DECAY);
    float nc = (cnt + LAMBDA) / ((float)NR + (float)KC * LAMBDA) * (float)NR;
    wsf[WS_NEWCOUNT + t] = nc;
    out[OUT_NEWCOUNT + t] = nc;

    float avg = (float)wsi[WS_HIST + t] / (float)NR;
    float term = avg * logf(avg + 1e-10f);
    for (int off = 16; off; off >>= 1) term += __shfl_xor(term, off, 32);
    if ((t & 31) == 0) red[t >> 5] = term;
    __syncthreads();
    if (t < 32) {
        float v = red[t];
        for (int off = 16; off; off >>= 1) v += __shfl_xor(v, off, 32);
        if (t == 0) {
            out[OUT_PERP] = expf(-v);
            out[OUT_LOSS] = COMMIT * wsf[WS_LOSS] / (float)NR;
        }
    }
}

// ---------------------------------------------------------------- d_weight GEMM (WMMA f32) + EMA epilogue
// block = 8 waves (8 k-tiles) sharing one 16-wide d tile; x streamed to LDS in
// 256-row chunks via async DMA. out tile [16 k x 16 d], reduction over N.
#define NCH  256
#define XPAD 257   // LDS row stride (dwords)
__global__ void __launch_bounds__(256) dweight_kernel(const float* __restrict__ soft,
                                                      const float* __restrict__ inp,
                                                      const float* __restrict__ ema_weight,
                                                      const float* __restrict__ wsf,
                                                      float* __restrict__ out) {
    __shared__ float xl[16 * XPAD];
    int t = threadIdx.x;
    int db = blockIdx.x & 15;  int d0 = db * 16;      // 16 d-blocks
    int kg = blockIdx.x >> 4;                         // 8 k-groups
    int lane = t & 31;
    int lo = lane & 15, hi = lane >> 4;
    int k0 = (kg * 8 + (t >> 5)) * 16;                // wave's k tile
    int kA = k0 + lo;
    int dloc = t >> 4;                                // 0..15 : d within tile
    int nseg = t & 15;                                // 0..15 : 16-row segment

    v8f c = {};
    for (int nc = 0; nc < NR; nc += NCH) {
        __syncthreads();                              // previous chunk consumed
        {
            const float* gsrc = inp + xoff(nc + nseg * 16, d0 + dloc);
            unsigned lbase = (unsigned)(unsigned long long)&xl[dloc * XPAD + nseg * 16];
#pragma unroll
            for (int j = 0; j < 4; ++j)
                async_g2l_b128(lbase + j * 16, gsrc + j * 4);
            wait_async0();
        }
        __syncthreads();
        for (int n = 0; n < NCH; n += 4) {
            int nl = nc + n + 2 * hi;                 // K-dim index = v + 2*hi
            int nrel = n + 2 * hi;
            v2f a, b;
            a.x = soft[(size_t)nl * KC + kA];
            a.y = soft[(size_t)(nl + 1) * KC + kA];
            b.x = xl[lo * XPAD + nrel];
            b.y = xl[lo * XPAD + nrel + 1];
            c = __builtin_amdgcn_wmma_f32_16x16x4_f32(false, a, false, b, (short)0, c, false, false);
        }
    }
#pragma unroll
    for (int r = 0; r < 8; ++r) {
        int k = k0 + r + 8 * hi;
        int d = d0 + lo;
        float ne = ema_weight[(size_t)k * DD + d] * DECAY + c[r] * (1.0f - DECAY);
        out[OUT_NEWEMA + (size_t)k * DD + d] = ne;
        out[OUT_NEWEMB + (size_t)k * DD + d] = ne / wsf[WS_NEWCOUNT + k];
    }
}

// ---------------------------------------------------------------- quantized gather -> [B,D,H,W]
__global__ void quant_kernel(const float* __restrict__ weight, const int* wsi,
                             float* __restrict__ out) {
    int n = blockIdx.x * blockDim.x + threadIdx.x;
    if (n >= NR) return;
    const float* w = weight + (size_t)wsi[WS_IDX + n] * DD;
    size_t base = OUT_QUANT + (size_t)(n >> 10) * (size_t)DHW + (size_t)(n & 1023);
    for (int d = 0; d < DD; ++d)
        out[base + (size_t)d * HW] = w[d];   // quantized + NU*(q - sg(q)) == quantized in value
}

// ---------------------------------------------------------------- one-hot overwrite of encodings
__global__ void onehot_kernel(const int* wsi, float* __restrict__ enc) {
    size_t g = (size_t)blockIdx.x * blockDim.x + threadIdx.x;
    int n = (int)(g >> 10);
    int k = (int)(g & 1023);
    enc[g] = (wsi[WS_IDX + n] == k) ? 1.0f : 0.0f;
}

extern "C" void kernel_launch(void* const* d_in, const int* in_sizes, int n_in,
                              void* d_out, int out_size, void* d_ws, size_t ws_size,
                              hipStream_t stream) {
    const float* inp        = (const float*)d_in[0];  // [B,D,H,W]
    const float* weight     = (const float*)d_in[1];  // [K,D]
    const float* ema_count  = (const float*)d_in[2];  // [K]
    const float* ema_weight = (const float*)d_in[3];  // [K,D]
    const float* gumbel     = (const float*)d_in[4];  // [N,K]
    float* out = (float*)d_out;
    float* wsf = (float*)d_ws;
    int*   wsi = (int*)d_ws;
    float* enc = out + OUT_ENC;                       // [N,K] scratch: dist -> soft -> one-hot

    zero_kernel    <<<4, 256, 0, stream>>>(wsf, wsi);
    norms_kernel   <<<(NR + 255) / 256, 256, 0, stream>>>(inp, weight, wsf);
    dist_kernel    <<<(NR / 128) * (KC / 64), 256, 0, stream>>>(inp, weight, wsf, enc);
    row_kernel     <<<NR / 8, 256, 0, stream>>>(enc, gumbel, wsf, wsi);
    finalize_kernel<<<1, 1024, 0, stream>>>(ema_count, wsf, wsi, out);
    dweight_kernel <<<(KC / 128) * (DD / 16) * 8 / 8 * 8 / 8 * 16 / 16 * 8, 256, 0, stream>>>(enc, inp, ema_weight, wsf, out);
    quant_kernel   <<<(NR + 255) / 256, 256, 0, stream>>>(weight, wsi, out);
    onehot_kernel  <<<(unsigned)(((size_t)NR * KC) / 256), 256, 0, stream>>>(wsi, enc);
}